// DifferentiableEmbedding_19782619365560
// MI455X (gfx1250) — compile-verified
//
#include <hip/hip_runtime.h>
#include <math.h>

// out[t, d] = emb[ids[t], d] * ( (d < g) + frac(1e9*g)/1e9 * tanh(g) ),
//   g = gates[ids[t]],  T = 65536 tokens, D = 256 f32.
//
// Memory-bound gather+scale: ~134 MB of traffic, ~0.05 GFLOP. On MI455X
// (23.3 TB/s) the floor is ~5.7 us; matrix cores cannot reduce bytes, so the
// optimal kernel is a wave32 row-gather with b128 loads (RT-cached: 128 MB
// table fits the 192 MB L2) and non-temporal b128 stores (write-once output
// must not evict hot embedding rows).

typedef float v4f __attribute__((ext_vector_type(4)));

#define D_DIM 256      // floats per row (fixed by the reference)
#define WAVES_PER_BLOCK 8

__global__ __launch_bounds__(256) void
diff_embedding_gate_kernel(const int*   __restrict__ ids,    // [T]
                           const float* __restrict__ emb,    // [V, 256]
                           const float* __restrict__ gates,  // [V]
                           float*       __restrict__ out,    // [T, 256]
                           int T)
{
    // One token per wave32. Token index is wave-uniform; readfirstlane tells
    // the compiler so, turning ids[t] / gates[idx] into scalar (SMEM) loads.
    const int lane = threadIdx.x & 31;
    int t = (int)(blockIdx.x * WAVES_PER_BLOCK) + (int)(threadIdx.x >> 5);
    t = __builtin_amdgcn_readfirstlane(t);
    if (t >= T) return;

    const int   idx = ids[t];
    const float g   = gates[idx];

    // Faithful f32 replication of the reference's soft term.
    const float gl   = g * 1e9f;
    const float soft = ((gl - floorf(gl)) / 1e9f) * tanhf(g);

    const v4f* __restrict__ row  = (const v4f*)(emb + (size_t)idx * D_DIM);
    v4f*       __restrict__ orow = (v4f*)(out + (size_t)t * D_DIM);

    // Lane l covers float4 slots {l, l+32}: two fully-coalesced 512 B
    // global_load_b128 bursts per wave (1 KB row). Issue both loads first.
    v4f v0 = row[lane];
    v4f v1 = row[lane + 32];

    const float d0 = (float)(lane * 4);         // element index of v0.x
    const float d1 = (float)(lane * 4 + 128);   // element index of v1.x

    v4f m0, m1;
    m0.x = ((d0 + 0.0f) < g ? 1.0f : 0.0f) + soft;
    m0.y = ((d0 + 1.0f) < g ? 1.0f : 0.0f) + soft;
    m0.z = ((d0 + 2.0f) < g ? 1.0f : 0.0f) + soft;
    m0.w = ((d0 + 3.0f) < g ? 1.0f : 0.0f) + soft;
    m1.x = ((d1 + 0.0f) < g ? 1.0f : 0.0f) + soft;
    m1.y = ((d1 + 1.0f) < g ? 1.0f : 0.0f) + soft;
    m1.z = ((d1 + 2.0f) < g ? 1.0f : 0.0f) + soft;
    m1.w = ((d1 + 3.0f) < g ? 1.0f : 0.0f) + soft;

    v0 *= m0;
    v1 *= m1;

    // Streaming write-once output: non-temporal (TH=NT) b128 stores so the
    // 64 MiB result doesn't flush the embedding working set out of L2.
    __builtin_nontemporal_store(v0, &orow[lane]);
    __builtin_nontemporal_store(v1, &orow[lane + 32]);
}

extern "C" void kernel_launch(void* const* d_in, const int* in_sizes, int n_in,
                              void* d_out, int out_size, void* d_ws, size_t ws_size,
                              hipStream_t stream)
{
    // setup_inputs() order: input_ids [B*S], embedding [V*D], gates [V]
    const int*   ids   = (const int*)d_in[0];
    const float* emb   = (const float*)d_in[1];
    const float* gates = (const float*)d_in[2];
    float*       out   = (float*)d_out;

    const int T = in_sizes[0];                       // 32 * 2048 = 65536 tokens
    const int blocks = (T + WAVES_PER_BLOCK - 1) / WAVES_PER_BLOCK;

    diff_embedding_gate_kernel<<<blocks, 256, 0, stream>>>(ids, emb, gates, out, T);
}